// NN2_14620068675687
// MI455X (gfx1250) — compile-verified
//
#include <hip/hip_runtime.h>
#include <hip/hip_bf16.h>

typedef __attribute__((ext_vector_type(2))) float v2f;
typedef __attribute__((ext_vector_type(8))) float v8f;
typedef unsigned long long ull;

#define DDIM 128
#define BM 128
#define BN 128
#define BK 32
// Pair-interleaved LDS: element (k,n) lives in float2 P[(k>>1)*S2 + n][k&1].
// S2 = 80 float2 per pair-row -> 160 dword stride; 160 mod 64 == 32, so the two
// half-wave pair-rows of a fragment load occupy disjoint bank halves.
#define S2 80

// Monotone packing: key(s) strictly increasing with float s; low 32 bits ~idx so that
// on score ties the LARGER packed value has the SMALLER index (matches jnp.argmax).
__device__ __forceinline__ ull packScoreIdx(float s, int idx) {
    unsigned u = __float_as_uint(s);
    unsigned key = (u & 0x80000000u) ? ~u : (u | 0x80000000u);
    return ((ull)key << 32) | (unsigned)(~(unsigned)idx);
}
__device__ __forceinline__ int unpackIdx(ull p) {
    return (int)~(unsigned)(p & 0xFFFFFFFFu);
}
__device__ __forceinline__ float unpackScore(ull p) {
    unsigned key = (unsigned)(p >> 32);
    unsigned u = (key & 0x80000000u) ? (key ^ 0x80000000u) : ~key;
    return __uint_as_float(u);
}

__global__ void nn2_init_kernel(ull* buf, int n) {
    int i = blockIdx.x * blockDim.x + threadIdx.x;
    if (i < n) buf[i] = 0ULL;   // 0 is below every packed finite score
}

__global__ __launch_bounds__(256)
void nn2_sim_kernel(const float* __restrict__ desc1, const float* __restrict__ desc2,
                    ull* __restrict__ rowBest, ull* __restrict__ colBest,
                    int N1, int N2) {
    __shared__ float2 Asp[(BK / 2) * S2];   // 16 pair-rows x 80 float2 = 10.0 KB
    __shared__ float2 Bsp[(BK / 2) * S2];
    __shared__ ull    colRed[BN];

    const int tid  = threadIdx.x;
    const int lane = tid & 31;
    const int wave = tid >> 5;        // 0..7
    const int half = lane >> 4;       // 0 or 1
    const int l15  = lane & 15;
    const int m0   = wave * 16;       // this wave's M strip within the block

    const int i0 = blockIdx.y * BM;   // global M (desc1 column) base
    const int j0 = blockIdx.x * BN;   // global N (desc2 column) base

    if (tid < BN) colRed[tid] = 0ULL;

    v8f acc[8];
#pragma unroll
    for (int t = 0; t < 8; ++t) acc[t] = (v8f){0.f,0.f,0.f,0.f,0.f,0.f,0.f,0.f};

    for (int kc = 0; kc < DDIM; kc += BK) {
        __syncthreads();
        // Cooperative fill, pair-interleaved. 1024 tasks per panel; task = (4 k's, 1 column n).
        // Lanes advance along n (contiguous in memory) -> coalesced global loads; the two
        // float2 LDS stores per task are b64, lanes covering all 64 banks exactly once.
#pragma unroll
        for (int it = 0; it < 4; ++it) {
            int idx = tid + it * 256;
            int n   = idx & 127;
            int k4  = (idx >> 7) << 2;            // 0,4,...,28
            const float* p1 = desc1 + (size_t)(kc + k4) * N1 + i0 + n;
            const float* p2 = desc2 + (size_t)(kc + k4) * N2 + j0 + n;
            float a0 = p1[0], a1 = p1[N1], a2 = p1[2 * (size_t)N1], a3 = p1[3 * (size_t)N1];
            float b0 = p2[0], b1 = p2[N2], b2 = p2[2 * (size_t)N2], b3 = p2[3 * (size_t)N2];
            int pr = k4 >> 1;                     // pair-row
            Asp[pr * S2 + n]       = make_float2(a0, a1);
            Asp[(pr + 1) * S2 + n] = make_float2(a2, a3);
            Bsp[pr * S2 + n]       = make_float2(b0, b1);
            Bsp[(pr + 1) * S2 + n] = make_float2(b2, b3);
        }
        // Prefetch next K chunk while we compute on this one.
        if (kc + BK < DDIM && tid < BK) {
            __builtin_prefetch(desc1 + (size_t)(kc + BK + tid) * N1 + i0, 0, 3);
            __builtin_prefetch(desc2 + (size_t)(kc + BK + tid) * N2 + j0, 0, 3);
        }
        __syncthreads();

#pragma unroll
        for (int k = 0; k < BK; k += 4) {
            // f32 16x16x4 WMMA operand layout: lanes 0-15 carry K=k,k+1 (pair-row k>>1),
            // lanes 16-31 carry K=k+2,k+3 (pair-row k>>1 + 1). One b64 LDS load per fragment.
            int kp = (k >> 1) + half;
            // Batch all fragment loads for this k-step first so one dscnt wait covers
            // eight back-to-back WMMA issues (instead of a wait per WMMA pair).
            v2f a = *(const v2f*)&Asp[kp * S2 + m0 + l15];
            v2f b[8];
#pragma unroll
            for (int t = 0; t < 8; ++t)
                b[t] = *(const v2f*)&Bsp[kp * S2 + t * 16 + l15];
#pragma unroll
            for (int t = 0; t < 8; ++t)
                acc[t] = __builtin_amdgcn_wmma_f32_16x16x4_f32(
                    /*neg_a=*/false, a, /*neg_b=*/false, b[t],
                    /*c_mod=*/(short)0, acc[t], /*reuse_a=*/false, /*reuse_b=*/false);
        }
    }

    // --- Row argmax (over N) ---
    // C/D layout: lane holds column N = l15, VGPR r holds row M = half*8 + r.
#pragma unroll
    for (int r = 0; r < 8; ++r) {
        float best = acc[0][r];
        int   bj   = j0 + l15;
#pragma unroll
        for (int t = 1; t < 8; ++t) {
            float v = acc[t][r];
            int   j = j0 + t * 16 + l15;
            if (v > best) { best = v; bj = j; }
        }
        ull p = packScoreIdx(best, bj);
#pragma unroll
        for (int mask = 8; mask >= 1; mask >>= 1) {
            ull o = __shfl_xor(p, mask, 16);   // butterfly within each 16-lane half
            if (o > p) p = o;
        }
        if (l15 == 0) {
            int m = i0 + m0 + half * 8 + r;
            atomicMax(&rowBest[m], p);
        }
    }

    // --- Column argmax (over M), pre-reduced in LDS across the 8 waves ---
#pragma unroll
    for (int t = 0; t < 8; ++t) {
        float best = acc[t][0];
        int   bm   = i0 + m0 + half * 8;
#pragma unroll
        for (int r = 1; r < 8; ++r) {
            float v = acc[t][r];
            int   m = i0 + m0 + half * 8 + r;
            if (v > best) { best = v; bm = m; }
        }
        ull p = packScoreIdx(best, bm);
        ull o = __shfl_xor(p, 16, 32);         // combine the two halves (same N, different M)
        if (o > p) p = o;
        if (half == 0) atomicMax(&colRed[t * 16 + l15], p);
    }
    __syncthreads();
    if (tid < BN) atomicMax(&colBest[j0 + tid], colRed[tid]);
}

__global__ void nn2_finalize_kernel(const ull* __restrict__ rowBest,
                                    const ull* __restrict__ colBest,
                                    int* __restrict__ outIdx,
                                    float* __restrict__ outScore, int N1) {
    int i = blockIdx.x * blockDim.x + threadIdx.x;
    if (i >= N1) return;
    ull rp = rowBest[i];
    int j = unpackIdx(rp);
    float s = unpackScore(rp);
    int i2 = unpackIdx(colBest[j]);
    bool mutual = (i2 == i);
    outIdx[i]   = mutual ? j : -1;
    outScore[i] = mutual ? s : -1.0f;
}

extern "C" void kernel_launch(void* const* d_in, const int* in_sizes, int n_in,
                              void* d_out, int out_size, void* d_ws, size_t ws_size,
                              hipStream_t stream) {
    const float* desc1 = (const float*)d_in[0];   // [D, N1]
    const float* desc2 = (const float*)d_in[1];   // [D, N2]
    // d_in[2], d_in[3] (kpts) are unused by the reference computation.

    const int N1 = in_sizes[0] / DDIM;
    const int N2 = in_sizes[1] / DDIM;

    ull* rowBest = (ull*)d_ws;
    ull* colBest = rowBest + N1;

    int nInit = N1 + N2;
    nn2_init_kernel<<<(nInit + 255) / 256, 256, 0, stream>>>(rowBest, nInit);

    dim3 grid(N2 / BN, N1 / BM);
    nn2_sim_kernel<<<grid, 256, 0, stream>>>(desc1, desc2, rowBest, colBest, N1, N2);

    int* outIdx = (int*)d_out;                     // indices0: int32 [1, N1]
    float* outScore = (float*)d_out + N1;          // mscores0: f32  [1, N1]
    nn2_finalize_kernel<<<(N1 + 255) / 256, 256, 0, stream>>>(rowBest, colBest, outIdx, outScore, N1);
}